// NearestNeighbourLoss_17910013624795
// MI455X (gfx1250) — compile-verified
//
#include <hip/hip_runtime.h>
#include <math.h>

// Problem constants (match reference)
#define B_     4
#define C_     256          // embed dim = WMMA K extent
#define HW_    4096         // 64*64
#define NC_    4
#define N_     2048         // past samples
#define NPIX_  (B_*HW_)     // 16384 pixels
#define NT_N_  (N_/16)      // 128 N-tiles
#define NT_M_  (NPIX_/16)   // 1024 M-tiles
#define EPS_   1e-8f
#define INVT_  10.0f        // 1/TEMP

typedef float v2f __attribute__((ext_vector_type(2)));
typedef float v8f __attribute__((ext_vector_type(8)));

// Paired-K fragment layout (both A-tile in LDS and B-tiles in workspace):
//   element (k, idx16) lives at float offset
//     ((k>>2)*32 + ((k>>1)&1)*16 + idx16)*2 + (k&1)
// so a lane's (k, k+1) pair for WMMA is one aligned 8-byte v2f, and the
// per-wmma fragment address is simply  base_v2f[kq*32 + lane].

// ---------------- init: zero class counts + global sum ----------------
__global__ void nn_init_kernel(int* ccount, float* sumAcc) {
    int t = threadIdx.x;
    if (t < NC_) ccount[t] = 0;
    if (t == 0)  sumAcc[0] = 0.0f;
}

// ------- normalize past samples into paired-K B-fragment layout -------
__global__ __launch_bounds__(C_)
void nn_normP_kernel(const float* __restrict__ P, const float* __restrict__ L,
                     float* __restrict__ PnW, int* __restrict__ cls,
                     int* __restrict__ ccount) {
    __shared__ float nrm2;
    int n = blockIdx.x;
    int t = threadIdx.x;                 // k index, 0..255
    float v = P[n * C_ + t];
    if (t == 0) nrm2 = 0.0f;
    __syncthreads();
    atomicAdd(&nrm2, v * v);             // ds_add_f32
    __syncthreads();
    float inv = 1.0f / fmaxf(sqrtf(nrm2), EPS_);
    size_t idx = ((((size_t)(n >> 4) * (C_ / 4) + (t >> 2)) * 32
                   + ((t >> 1) & 1) * 16 + (n & 15)) << 1) + (t & 1);
    PnW[idx] = v * inv;
    if (t == 0) {
        int c = 0; float best = L[n * NC_];
        #pragma unroll
        for (int k = 1; k < NC_; ++k) {
            float val = L[n * NC_ + k];
            if (val > best) { best = val; c = k; }
        }
        cls[n] = c;
        atomicAdd(&ccount[c], 1);
    }
}

// ---------------- per-pixel class from one-hot y ----------------------
__global__ void nn_pcls_kernel(const float* __restrict__ y, int* __restrict__ pcls) {
    int i = blockIdx.x * blockDim.x + threadIdx.x;    // 0..NPIX_-1
    if (i >= NPIX_) return;
    int b = i >> 12, v = i & (HW_ - 1);
    int c = 0; float best = y[((b * NC_ + 0) << 12) + v];
    #pragma unroll
    for (int k = 1; k < NC_; ++k) {
        float val = y[((b * NC_ + k) << 12) + v];
        if (val > best) { best = val; c = k; }
    }
    pcls[i] = c;
}

// ---------------- fused WMMA GEMM + exp + ratio -----------------------
// One block per 16-pixel M-tile. 8 wave32s; each wave covers 16 N-tiles.
__global__ __launch_bounds__(256)
void nn_main_kernel(const float* __restrict__ X, const float* __restrict__ PnW,
                    const int* __restrict__ cls, const int* __restrict__ ccount,
                    const int* __restrict__ pcls, float* __restrict__ sumAcc) {
    __shared__ float sX[C_ * 16];        // X tile, paired-K layout (16 KB)
    __shared__ float nrm2[16];
    __shared__ float accA[16], accS[16];
    __shared__ int   pc[16];

    const int t  = threadIdx.x;
    const int mt = blockIdx.x;
    const int m  = t & 15;               // pixel within tile
    const int kk = t >> 4;               // 0..15 -> handles k = kk + 16*j
    const int p  = mt * 16 + m;          // global pixel
    const int b  = p >> 12;
    const int v  = p & (HW_ - 1);

    // stage raw X (coalesced: consecutive lanes -> consecutive v), partial norms
    float vals[16];
    float part = 0.0f;
    #pragma unroll
    for (int j = 0; j < 16; ++j) {
        int k = kk + j * 16;
        float x = X[((b * C_ + k) << 12) + v];
        vals[j] = x;
        part += x * x;
    }
    if (t < 16) { nrm2[t] = 0.0f; accA[t] = 0.0f; accS[t] = 0.0f; pc[t] = pcls[mt * 16 + t]; }
    __syncthreads();
    atomicAdd(&nrm2[m], part);           // ds_add_f32
    __syncthreads();
    float inv = 1.0f / fmaxf(sqrtf(nrm2[m]), EPS_);
    #pragma unroll
    for (int j = 0; j < 16; ++j) {
        int k = kk + j * 16;
        sX[(((k >> 2) * 32 + ((k >> 1) & 1) * 16 + m) << 1) + (k & 1)] = vals[j] * inv;
    }
    __syncthreads();

    const int wave = t >> 5;             // 0..7
    const int lane = t & 31;
    const int half = lane >> 4;
    const int ln   = lane & 15;          // A: M index; B/C: N index

    float aAll[8], aSame[8];
    #pragma unroll
    for (int r = 0; r < 8; ++r) { aAll[r] = 0.0f; aSame[r] = 0.0f; }

    const v2f* aPtr = (const v2f*)sX;    // fragment = aPtr[kq*32 + lane]

    for (int nt = wave; nt < NT_N_; nt += 8) {
        const int  nglob = nt * 16 + ln;
        const int  scls  = cls[nglob];
        const v2f* bPtr  = (const v2f*)(PnW + (size_t)nt * C_ * 16);

        v8f c = {0.f, 0.f, 0.f, 0.f, 0.f, 0.f, 0.f, 0.f};
        #pragma unroll 8
        for (int kq = 0; kq < C_ / 4; ++kq) {
            v2f a  = aPtr[kq * 32 + lane];   // ds_load_b64
            v2f bb = bPtr[kq * 32 + lane];   // global_load_b64, coalesced 256B/wave
            // D = A(16x4 f32) * B(4x16 f32) + C(16x16 f32)
            c = __builtin_amdgcn_wmma_f32_16x16x4_f32(
                    false, a, false, bb, (short)0, c, false, false);
        }
        #pragma unroll
        for (int r = 0; r < 8; ++r) {
            float sim = c[r];                         // (M=r+8*half, N=ln)
            float e   = __expf((sim - 1.0f) * INVT_); // exp(-dist/T)
            aAll[r] += e;
            if (scls == pc[r + 8 * half]) aSame[r] += e;
        }
    }

    #pragma unroll
    for (int r = 0; r < 8; ++r) {
        atomicAdd(&accA[r + 8 * half], aAll[r]);     // ds_add_f32
        atomicAdd(&accS[r + 8 * half], aSame[r]);
    }
    __syncthreads();

    if (t < 16) {
        int   k   = pc[t];
        float cnt = (float)ccount[k];
        float num = accS[t] + ((float)N_ - cnt);     // non-matching "same" terms are exp(0)=1
        float den = cnt + (accA[t] - accS[t]);       // matching "other" terms are 1
        atomicAdd(sumAcc, num / den);                // global_atomic_add_f32
    }
}

// ---------------- finalize ----------------
__global__ void nn_fin_kernel(const float* sumAcc, float* out) {
    out[0] = sumAcc[0] * (1.0f / (float)NPIX_);
}

extern "C" void kernel_launch(void* const* d_in, const int* in_sizes, int n_in,
                              void* d_out, int out_size, void* d_ws, size_t ws_size,
                              hipStream_t stream) {
    const float* X = (const float*)d_in[0];   // (B,C,H,W)
    const float* y = (const float*)d_in[1];   // (B,NC,H,W)
    const float* P = (const float*)d_in[2];   // (N,C)
    const float* L = (const float*)d_in[3];   // (N,NC)
    float* out = (float*)d_out;

    // workspace layout
    float* PnW    = (float*)d_ws;             // N_*C_ floats, paired-K tile layout
    int*   cls    = (int*)(PnW + (size_t)N_ * C_);
    int*   ccount = cls + N_;
    int*   pcls   = ccount + NC_;
    float* sumAcc = (float*)(pcls + NPIX_);

    nn_init_kernel<<<1, 32, 0, stream>>>(ccount, sumAcc);
    nn_normP_kernel<<<N_, C_, 0, stream>>>(P, L, PnW, cls, ccount);
    nn_pcls_kernel<<<NPIX_ / 256, 256, 0, stream>>>(y, pcls);
    nn_main_kernel<<<NT_M_, 256, 0, stream>>>(X, PnW, cls, ccount, pcls, sumAcc);
    nn_fin_kernel<<<1, 1, 0, stream>>>(sumAcc, out);
}